// RNNTLoss_16226386444947
// MI455X (gfx1250) — compile-verified
//
#include <hip/hip_runtime.h>
#include <hip/hip_bf16.h>

// RNNT loss, B=8 T=256 U=64 (U1=65) V=1024.
// Phase 1: bandwidth-bound per-row logsumexp (one wave32 per 1024-float row).
// Phase 2: C = cumsum_u(Lpad) as batched triangular GEMM via V_WMMA_F32_16X16X4_F32.
// Phase 3: sequential t-scan with wave-level stabilized prefix-logsumexp (one wave per batch).
// Phase 4: mean of per-batch costs.

#define RB 8
#define RT 256
#define RU 64
#define RU1 65
#define RV 1024
#define CSTRIDE 80   // padded C row (5 u-tiles of 16)

typedef __attribute__((ext_vector_type(2))) float v2f;
typedef __attribute__((ext_vector_type(8))) float v8f;

__device__ __forceinline__ float wave_max(float v) {
#pragma unroll
  for (int off = 16; off > 0; off >>= 1)
    v = fmaxf(v, __shfl_xor(v, off, 32));
  return v;
}
__device__ __forceinline__ float wave_sum(float v) {
#pragma unroll
  for (int off = 16; off > 0; off >>= 1)
    v += __shfl_xor(v, off, 32);
  return v;
}

// ---------------------------------------------------------------------------
// Phase 1: per-row stats. One wave per (b,t,u) row of V=1024 floats.
// Writes blank_lp[row] and Lpad[(b*T+t)*65 + u'] (label lp shifted by 1, col 0 = 0).
// ---------------------------------------------------------------------------
__global__ __launch_bounds__(256) void rnnt_row_stats(
    const float* __restrict__ logits, const int* __restrict__ targets,
    float* __restrict__ blankbuf, float* __restrict__ lpadbuf, int nrows) {
  const int lane = threadIdx.x & 31;
  const int wave = threadIdx.x >> 5;
  const int row = blockIdx.x * 8 + wave;
  if (row >= nrows) return;

  const int b = row / (RT * RU1);
  const int t = (row / RU1) % RT;
  const int u = row % RU1;

  const size_t base = (size_t)row * RV;
  const float4* p = reinterpret_cast<const float4*>(logits + base);

  // Load the whole row into registers: 8 float4 per lane (256 float4 per row).
  float4 r[8];
#pragma unroll
  for (int i = 0; i < 8; ++i) r[i] = p[i * 32 + lane];

  // Max.
  float m = -INFINITY;
#pragma unroll
  for (int i = 0; i < 8; ++i) {
    m = fmaxf(m, fmaxf(fmaxf(r[i].x, r[i].y), fmaxf(r[i].z, r[i].w)));
  }
  m = wave_max(m);

  // Sum of exp(v - m).
  float s = 0.f;
#pragma unroll
  for (int i = 0; i < 8; ++i) {
    s += __expf(r[i].x - m) + __expf(r[i].y - m) +
         __expf(r[i].z - m) + __expf(r[i].w - m);
  }
  s = wave_sum(s);
  const float lse = m + __logf(s);

  if (lane == 0) {
    const float blank = logits[base] - lse;     // lp[b,t,u,0]
    blankbuf[row] = blank;
    const int bt = b * RT + t;
    if (u < RU) {
      const int tgt = targets[b * RU + u];
      lpadbuf[bt * RU1 + (u + 1)] = logits[base + tgt] - lse;
    }
    if (u == 0) lpadbuf[bt * RU1 + 0] = 0.f;
  }
}

// ---------------------------------------------------------------------------
// Phase 2: C[b,t,u] = sum_{v<=u} Lpad[b,t,v]  ==  Lpad(256x68) @ Ltri^T(68x65)
// per batch, done as 16x16 f32 WMMA tiles.  One wave per (b, t-tile).
// A (16x4 f32) layout: lanes 0-15 -> M=lane, K=4j..4j+1; lanes 16-31 -> K=4j+2..4j+3.
// B (4x16 f32) layout assumed: VGPR0 -> K= (lane<16?0:1), N=lane%16; VGPR1 -> K=(lane<16?2:3).
// D (16x16 f32) layout (documented): VGPR r -> M = r + 8*(lane>=16), N = lane%16.
// ---------------------------------------------------------------------------
__global__ __launch_bounds__(32) void rnnt_cumsum_wmma(
    const float* __restrict__ lpadbuf, float* __restrict__ cbuf) {
  const int lane = threadIdx.x & 31;
  const int hi = lane >> 4;          // 0: lanes 0-15, 1: lanes 16-31
  const int nn = lane & 15;
  const int b = blockIdx.x >> 4;     // 16 t-tiles per batch
  const int ttile = blockIdx.x & 15;

  const int trow = ttile * 16 + nn;  // M row this lane feeds into A
  const float* lrow = lpadbuf + (size_t)(b * RT + trow) * RU1;

  for (int n = 0; n < 5; ++n) {          // u-tiles: cols n*16 .. n*16+15
    v8f acc = {};
    const int uglob = n * 16 + nn;       // this lane's B column
#pragma unroll
    for (int j = 0; j < 17; ++j) {       // K = 4j .. 4j+3, K<68
      const int ka = 4 * j + 2 * hi;     // A element pair for this lane
      v2f a;
      a.x = (ka + 0 < RU1) ? lrow[ka + 0] : 0.f;
      a.y = (ka + 1 < RU1) ? lrow[ka + 1] : 0.f;
      const int kb0 = 4 * j + (hi ? 1 : 0);
      const int kb1 = 4 * j + (hi ? 3 : 2);
      v2f bm;
      bm.x = (kb0 <= uglob) ? 1.f : 0.f; // lower-triangular ones (transposed)
      bm.y = (kb1 <= uglob) ? 1.f : 0.f;
      acc = __builtin_amdgcn_wmma_f32_16x16x4_f32(
          /*neg_a=*/false, a, /*neg_b=*/false, bm,
          /*c_mod=*/(short)0, acc, /*reuse_a=*/false, /*reuse_b=*/false);
    }
#pragma unroll
    for (int r = 0; r < 8; ++r) {
      const int M = r + 8 * hi;
      const int tg = ttile * 16 + M;
      cbuf[(size_t)(b * RT + tg) * CSTRIDE + n * 16 + nn] = acc[r];
    }
  }
}

// ---------------------------------------------------------------------------
// Phase 3: alpha scan over t.  One wave per batch b.  Lane i owns u = {2i, 2i+1},
// lane 31 also owns u = 64.  Per step: stabilized prefix-logsumexp over 65 elems.
// ---------------------------------------------------------------------------
__global__ __launch_bounds__(32) void rnnt_scan(
    const float* __restrict__ blankbuf, const float* __restrict__ cbuf,
    const int* __restrict__ logit_lengths, const int* __restrict__ target_lengths,
    float* __restrict__ costs) {
  const int lane = threadIdx.x & 31;
  const int b = blockIdx.x;
  const int u0 = 2 * lane, u1 = 2 * lane + 1;
  const bool own3 = (lane == 31);        // u2 = 64
  const int tlast = logit_lengths[b] - 1;
  const int ulast = target_lengths[b];

  const float* Crow = cbuf + (size_t)(b * RT) * CSTRIDE;
  const float* Brow = blankbuf + (size_t)(b * RT) * RU1;

  float a0 = Crow[u0];
  float a1 = Crow[u1];
  float a2 = own3 ? Crow[64] : 0.f;      // alpha0 = C[:,0,:]

  float caught = 0.f;
  if (tlast == 0) {
    if (u0 == ulast) caught = a0 + Brow[u0];
    else if (u1 == ulast) caught = a1 + Brow[u1];
    else if (own3 && ulast == 64) caught = a2 + Brow[64];
  }

  for (int t = 1; t < RT; ++t) {
    const float* Ct = Crow + (size_t)t * CSTRIDE;
    const float* Bp = Brow + (size_t)(t - 1) * RU1;
    const float c0 = Ct[u0], c1 = Ct[u1];
    const float c2 = own3 ? Ct[64] : 0.f;

    const float x0 = a0 + Bp[u0] - c0;
    const float x1 = a1 + Bp[u1] - c1;
    const float x2 = own3 ? (a2 + Bp[64] - c2) : -INFINITY;

    float m = fmaxf(fmaxf(x0, x1), x2);
    m = wave_max(m);

    const float e0 = __expf(x0 - m);
    const float e1 = __expf(x1 - m);
    const float e2 = own3 ? __expf(x2 - m) : 0.f;

    const float p0 = e0;
    const float p1 = e0 + e1;
    const float p2 = p1 + e2;
    const float tot = own3 ? p2 : p1;

    // inclusive wave scan of lane totals
    float incl = tot;
#pragma unroll
    for (int d = 1; d < 32; d <<= 1) {
      float v = __shfl_up(incl, d, 32);
      if (lane >= d) incl += v;
    }
    const float excl = incl - tot;

    a0 = c0 + m + __logf(excl + p0);
    a1 = c1 + m + __logf(excl + p1);
    if (own3) a2 = c2 + m + __logf(excl + p2);

    if (t == tlast) {
      const float* Bt = Brow + (size_t)t * RU1;
      if (u0 == ulast) caught = a0 + Bt[u0];
      else if (u1 == ulast) caught = a1 + Bt[u1];
      else if (own3 && ulast == 64) caught = a2 + Bt[64];
    }
  }

  const float val = wave_sum(caught);    // exactly one lane nonzero
  if (lane == 0) costs[b] = -val;
}

// ---------------------------------------------------------------------------
// Phase 4: mean of per-batch costs -> d_out[0]
// ---------------------------------------------------------------------------
__global__ __launch_bounds__(32) void rnnt_mean(const float* __restrict__ costs,
                                                float* __restrict__ out) {
  const int lane = threadIdx.x & 31;
  float v = (lane < RB) ? costs[lane] : 0.f;
  v = wave_sum(v);
  if (lane == 0) out[0] = v / (float)RB;
}

extern "C" void kernel_launch(void* const* d_in, const int* in_sizes, int n_in,
                              void* d_out, int out_size, void* d_ws, size_t ws_size,
                              hipStream_t stream) {
  const float* logits = (const float*)d_in[0];
  const int* targets = (const int*)d_in[1];
  const int* logit_lengths = (const int*)d_in[2];
  const int* target_lengths = (const int*)d_in[3];
  float* out = (float*)d_out;

  float* ws = (float*)d_ws;
  const int nrows = RB * RT * RU1;             // 133120
  float* blankbuf = ws;                        // [B,T,65]
  float* lpadbuf = ws + nrows;                 // [B,T,65]
  float* cbuf = ws + 2 * nrows;                // [B,T,80] padded
  float* costs = cbuf + RB * RT * CSTRIDE;     // [B]

  // Phase 1: 8 waves per 256-thread block, one wave per row.
  rnnt_row_stats<<<dim3((nrows + 7) / 8), dim3(256), 0, stream>>>(
      logits, targets, blankbuf, lpadbuf, nrows);

  // Phase 2: one wave per (b, t-tile of 16): 8*16 = 128 blocks.
  rnnt_cumsum_wmma<<<dim3(RB * (RT / 16)), dim3(32), 0, stream>>>(lpadbuf, cbuf);

  // Phase 3: one wave per batch.
  rnnt_scan<<<dim3(RB), dim3(32), 0, stream>>>(blankbuf, cbuf, logit_lengths,
                                               target_lengths, costs);

  // Phase 4: reduce to scalar mean.
  rnnt_mean<<<dim3(1), dim3(32), 0, stream>>>(costs, out);
}